// MHCNModel_49512382988732
// MI455X (gfx1250) — compile-verified
//
#include <hip/hip_runtime.h>
#include <hip/hip_bf16.h>
#include <math.h>
#include <utility>

#define USER_NUM 100000
#define ITEM_NUM 40000
#define DIM      64
#define NNZ_S    2000000
#define NNZ_J    2000000
#define NNZ_P    1000000
#define NNZ_R    2000000

typedef float v2f __attribute__((ext_vector_type(2)));
typedef float v8f __attribute__((ext_vector_type(8)));

// ---------------------------------------------------------------------------
// Gating: out = emb * sigmoid(emb @ W^T + b), dual-written to out and acc.
// One wave computes a 16-row x 64-col output stripe with V_WMMA_F32_16X16X4_F32.
// A fragment (16x4 f32): lanes 0-15 hold M=lane, K={k0,k0+1}; lanes 16-31 hold
// M=lane-16, K={k0+2,k0+3}  -> one float2 load per lane per K-chunk.
// B fragment (4x16, B[k][n] = W[n][k]): same half-lane K split, N = lane&15.
// C/D (16x16 f32): VGPR v, lanes 0-15 -> M=v,N=lane; lanes 16-31 -> M=v+8.
// ---------------------------------------------------------------------------
__global__ __launch_bounds__(256) void gating_wmma_kernel(
    const float* __restrict__ emb, const float* __restrict__ W,
    const float* __restrict__ bias, float* __restrict__ out,
    float* __restrict__ acc, int nrows)
{
  __shared__ float Wl[64 * 65];                 // 65-float pitch: kill bank conflicts
  for (int i = threadIdx.x; i < 64 * 64; i += 256) {
    Wl[(i >> 6) * 65 + (i & 63)] = W[i];
  }
  __syncthreads();

  const int lane  = threadIdx.x & 31;
  const int wave  = threadIdx.x >> 5;
  const int rb    = (blockIdx.x * 8 + wave) * 16;
  if (rb >= nrows) return;                      // wave-uniform: EXEC stays all-ones

  const int m16   = lane & 15;
  const int khalf = lane >> 4;                  // 0 or 1
  const int r     = rb + m16;

  v8f accv[4] = {v8f{}, v8f{}, v8f{}, v8f{}};

  for (int k0 = 0; k0 < 64; k0 += 4) {
    const int kk = k0 + 2 * khalf;              // even -> 8B aligned float2
    const v2f a = *(const v2f*)(emb + r * 64 + kk);
#pragma unroll
    for (int t = 0; t < 4; ++t) {
      const int o = t * 16 + m16;               // output column this lane covers
      v2f b;
      b.x = Wl[o * 65 + kk];
      b.y = Wl[o * 65 + kk + 1];
      accv[t] = __builtin_amdgcn_wmma_f32_16x16x4_f32(
          /*neg_a=*/false, a, /*neg_b=*/false, b,
          /*c_mod=*/(short)0, accv[t], /*reuse_a=*/false, /*reuse_b=*/false);
    }
  }

  const int mbase = khalf * 8;
#pragma unroll
  for (int t = 0; t < 4; ++t) {
    const int n  = t * 16 + m16;
    const float bn = bias[n];
#pragma unroll
    for (int v = 0; v < 8; ++v) {
      const int row = rb + mbase + v;
      const float y = accv[t][v] + bn;
      const float s = 1.0f / (1.0f + expf(-y));
      const float g = emb[row * 64 + n] * s;
      out[row * 64 + n] = g;
      acc[row * 64 + n] = g;
    }
  }
}

// ---------------------------------------------------------------------------
// m[d] = sum_e att_mat[d][e] * att_vec[e]   (attention reduces to dot with m)
// ---------------------------------------------------------------------------
__global__ void compute_m_kernel(const float* __restrict__ att_mat,
                                 const float* __restrict__ att_vec,
                                 float* __restrict__ m)
{
  const int d = threadIdx.x;
  float s = 0.0f;
  for (int e = 0; e < 64; ++e) s += att_mat[d * 64 + e] * att_vec[e];
  m[d] = s;
}

__device__ __forceinline__ float wave_sum32(float v) {
#pragma unroll
  for (int mask = 16; mask >= 1; mask >>= 1) v += __shfl_xor(v, mask, 32);
  return v;
}

// ---------------------------------------------------------------------------
// mixed[n] = softmax_k(dot(e_k[n], m)) . e_k[n] + simple[n]/2 ; wave per row
// ---------------------------------------------------------------------------
__global__ __launch_bounds__(256) void attn_mix_kernel(
    const float* __restrict__ e1, const float* __restrict__ e2,
    const float* __restrict__ e3, const float* __restrict__ simple,
    const float* __restrict__ mvec, float* __restrict__ out, int nrows)
{
  const int lane = threadIdx.x & 31;
  const int n    = blockIdx.x * 8 + (threadIdx.x >> 5);
  if (n >= nrows) return;
  const long base = (long)n * 64;

  const float ma = mvec[lane], mb = mvec[lane + 32];
  const float x1a = e1[base + lane], x1b = e1[base + 32 + lane];
  const float x2a = e2[base + lane], x2b = e2[base + 32 + lane];
  const float x3a = e3[base + lane], x3b = e3[base + 32 + lane];

  const float w1 = wave_sum32(x1a * ma + x1b * mb);
  const float w2 = wave_sum32(x2a * ma + x2b * mb);
  const float w3 = wave_sum32(x3a * ma + x3b * mb);

  const float mx = fmaxf(w1, fmaxf(w2, w3));
  const float s1 = expf(w1 - mx), s2 = expf(w2 - mx), s3 = expf(w3 - mx);
  const float inv = 1.0f / (s1 + s2 + s3);

  out[base + lane] =
      (s1 * x1a + s2 * x2a + s3 * x3a) * inv + 0.5f * simple[base + lane];
  out[base + 32 + lane] =
      (s1 * x1b + s2 * x2b + s3 * x3b) * inv + 0.5f * simple[base + 32 + lane];
}

// ---------------------------------------------------------------------------
// acc[n] += x[n] / max(||x[n]||_2, 1e-12) ; wave per row
// ---------------------------------------------------------------------------
__global__ __launch_bounds__(256) void l2norm_acc_kernel(
    float* __restrict__ acc, const float* __restrict__ x, int nrows)
{
  const int lane = threadIdx.x & 31;
  const int n    = blockIdx.x * 8 + (threadIdx.x >> 5);
  if (n >= nrows) return;
  const long base = (long)n * 64;

  const float a = x[base + lane], b = x[base + 32 + lane];
  const float ss  = wave_sum32(a * a + b * b);
  const float inv = 1.0f / fmaxf(sqrtf(ss), 1e-12f);
  acc[base + lane]      += a * inv;
  acc[base + 32 + lane] += b * inv;
}

// ---------------------------------------------------------------------------
// COO SpMM scatter: out[rows[e]] += vals[e] * x[cols[e]] ; 16 lanes per nnz,
// each lane owns a float4 slice -> coalesced 128b gather + 4 f32 atomics.
// ---------------------------------------------------------------------------
__global__ __launch_bounds__(256) void spmm_atomic_kernel(
    const int* __restrict__ rows, const int* __restrict__ cols,
    const float* __restrict__ vals, const float* __restrict__ x,
    float* __restrict__ out, int nnz)
{
  const int gid = blockIdx.x * 256 + threadIdx.x;
  const int e   = gid >> 4;
  if (e >= nnz) return;
  const int sub = (gid & 15) * 4;

  const int   rr = rows[e];
  const int   cc = cols[e];
  const float v  = vals[e];

  const float4 s = *(const float4*)(x + (long)cc * 64 + sub);
  float* d = out + (long)rr * 64 + sub;
  atomicAdd(d + 0, v * s.x);
  atomicAdd(d + 1, v * s.y);
  atomicAdd(d + 2, v * s.z);
  atomicAdd(d + 3, v * s.w);
}

__global__ __launch_bounds__(256) void zero4_kernel(float4* __restrict__ p, int n4) {
  const int i = blockIdx.x * 256 + threadIdx.x;
  if (i < n4) p[i] = float4{0.f, 0.f, 0.f, 0.f};
}

__global__ __launch_bounds__(256) void copy4_kernel(float4* __restrict__ d,
                                                    const float4* __restrict__ s, int n4) {
  const int i = blockIdx.x * 256 + threadIdx.x;
  if (i < n4) d[i] = s[i];
}

// ---------------------------------------------------------------------------
extern "C" void kernel_launch(void* const* d_in, const int* in_sizes, int n_in,
                              void* d_out, int out_size, void* d_ws, size_t ws_size,
                              hipStream_t stream) {
  const float* user_emb = (const float*)d_in[0];
  const float* item_emb = (const float*)d_in[1];
  const float* gc1_w = (const float*)d_in[2];  const float* gc1_b = (const float*)d_in[3];
  const float* gc2_w = (const float*)d_in[4];  const float* gc2_b = (const float*)d_in[5];
  const float* gc3_w = (const float*)d_in[6];  const float* gc3_b = (const float*)d_in[7];
  const float* gs_w  = (const float*)d_in[8];  const float* gs_b  = (const float*)d_in[9];
  const float* att_mat = (const float*)d_in[10];
  const float* att_vec = (const float*)d_in[11];
  const float* hs_vals = (const float*)d_in[12];
  const float* hj_vals = (const float*)d_in[13];
  const float* hp_vals = (const float*)d_in[14];
  const float* r_vals  = (const float*)d_in[15];
  const int* hs_rows = (const int*)d_in[16]; const int* hs_cols = (const int*)d_in[17];
  const int* hj_rows = (const int*)d_in[18]; const int* hj_cols = (const int*)d_in[19];
  const int* hp_rows = (const int*)d_in[20]; const int* hp_cols = (const int*)d_in[21];
  const int* r_rows  = (const int*)d_in[22]; const int* r_cols  = (const int*)d_in[23];

  const long UD = (long)USER_NUM * DIM;       // 6,400,000
  const long ID = (long)ITEM_NUM * DIM;       // 2,560,000

  float* out_user = (float*)d_out;            // user_all
  float* acc_i    = out_user + UD;            // acc_i lives in d_out

  float* ws = (float*)d_ws;
  float* uc1    = ws + 0 * UD;
  float* uc2    = ws + 1 * UD;
  float* uc3    = ws + 2 * UD;
  float* simple = ws + 3 * UD;
  float* acc1   = ws + 4 * UD;
  float* acc2   = ws + 5 * UD;
  float* acc3   = ws + 6 * UD;
  float* accS   = ws + 7 * UD;
  float* mixed  = ws + 8 * UD;
  float* tmp    = ws + 9 * UD;
  float* itemsA = ws + 10 * UD;
  float* itemsB = itemsA + ID;
  float* mvec   = itemsB + ID;

  const int gatingBlocks = (USER_NUM + 127) / 128;      // 8 waves x 16 rows
  const int rowBlocksU   = USER_NUM / 8;                // wave-per-row kernels
  const int rowBlocksI   = ITEM_NUM / 8;
  const int zUD = (int)(UD / 4 + 255) / 256, zID = (int)(ID / 4 + 255) / 256;

  // --- gating (WMMA) + accumulator init -----------------------------------
  gating_wmma_kernel<<<gatingBlocks, 256, 0, stream>>>(user_emb, gc1_w, gc1_b, uc1, acc1, USER_NUM);
  gating_wmma_kernel<<<gatingBlocks, 256, 0, stream>>>(user_emb, gc2_w, gc2_b, uc2, acc2, USER_NUM);
  gating_wmma_kernel<<<gatingBlocks, 256, 0, stream>>>(user_emb, gc3_w, gc3_b, uc3, acc3, USER_NUM);
  gating_wmma_kernel<<<gatingBlocks, 256, 0, stream>>>(user_emb, gs_w,  gs_b,  simple, accS, USER_NUM);
  compute_m_kernel<<<1, 64, 0, stream>>>(att_mat, att_vec, mvec);
  copy4_kernel<<<zID, 256, 0, stream>>>((float4*)itemsA, (const float4*)item_emb, (int)(ID / 4));
  copy4_kernel<<<zID, 256, 0, stream>>>((float4*)acc_i,  (const float4*)item_emb, (int)(ID / 4));

  float* items  = itemsA;
  float* nitems = itemsB;

  for (int layer = 0; layer < 2; ++layer) {
    // mixed = attention(uc1,uc2,uc3) + simple/2   (before uc/simple update)
    attn_mix_kernel<<<rowBlocksU, 256, 0, stream>>>(uc1, uc2, uc3, simple, mvec, mixed, USER_NUM);

    // uc1 = Hs @ uc1 ; acc1 += l2norm(uc1)
    zero4_kernel<<<zUD, 256, 0, stream>>>((float4*)tmp, (int)(UD / 4));
    spmm_atomic_kernel<<<(NNZ_S * 16 + 255) / 256, 256, 0, stream>>>(hs_rows, hs_cols, hs_vals, uc1, tmp, NNZ_S);
    std::swap(uc1, tmp);
    l2norm_acc_kernel<<<rowBlocksU, 256, 0, stream>>>(acc1, uc1, USER_NUM);

    // uc2 = Hj @ uc2
    zero4_kernel<<<zUD, 256, 0, stream>>>((float4*)tmp, (int)(UD / 4));
    spmm_atomic_kernel<<<(NNZ_J * 16 + 255) / 256, 256, 0, stream>>>(hj_rows, hj_cols, hj_vals, uc2, tmp, NNZ_J);
    std::swap(uc2, tmp);
    l2norm_acc_kernel<<<rowBlocksU, 256, 0, stream>>>(acc2, uc2, USER_NUM);

    // uc3 = Hp @ uc3
    zero4_kernel<<<zUD, 256, 0, stream>>>((float4*)tmp, (int)(UD / 4));
    spmm_atomic_kernel<<<(NNZ_P * 16 + 255) / 256, 256, 0, stream>>>(hp_rows, hp_cols, hp_vals, uc3, tmp, NNZ_P);
    std::swap(uc3, tmp);
    l2norm_acc_kernel<<<rowBlocksU, 256, 0, stream>>>(acc3, uc3, USER_NUM);

    // new_items = R^T @ mixed ; acc_i += l2norm(new_items)
    zero4_kernel<<<zID, 256, 0, stream>>>((float4*)nitems, (int)(ID / 4));
    spmm_atomic_kernel<<<(NNZ_R * 16 + 255) / 256, 256, 0, stream>>>(r_cols, r_rows, r_vals, mixed, nitems, NNZ_R);
    l2norm_acc_kernel<<<rowBlocksI, 256, 0, stream>>>(acc_i, nitems, ITEM_NUM);

    // simple = R @ items(old) ; accS += l2norm(simple)
    zero4_kernel<<<zUD, 256, 0, stream>>>((float4*)tmp, (int)(UD / 4));
    spmm_atomic_kernel<<<(NNZ_R * 16 + 255) / 256, 256, 0, stream>>>(r_rows, r_cols, r_vals, items, tmp, NNZ_R);
    std::swap(simple, tmp);
    l2norm_acc_kernel<<<rowBlocksU, 256, 0, stream>>>(accS, simple, USER_NUM);

    std::swap(items, nitems);
  }

  // user_all = attention(acc1,acc2,acc3) + accS/2
  attn_mix_kernel<<<rowBlocksU, 256, 0, stream>>>(acc1, acc2, acc3, accS, mvec, out_user, USER_NUM);
}